// DecoderRNN_31568009626349
// MI455X (gfx1250) — compile-verified
//
#include <hip/hip_runtime.h>
#include <hip/hip_bf16.h>
#include <math.h>

typedef __attribute__((ext_vector_type(2))) float v2f;
typedef __attribute__((ext_vector_type(8))) float v8f;

#define HIDDEN 2048
#define VOCAB  50257
#define WAVES_PER_BLOCK 8

// One wave computes a 16-row x 2048 f32 GEMV tile with V_WMMA_F32_16X16X4_F32.
// B operand (SRC1): lane n (n = lane&15) carries 2 consecutive f32 of weight
// row (rowbase+n); half-waves split the K=4 range (K{0,1} lanes 0-15,
// K{2,3} lanes 16-31), mirroring the documented 32-bit A layout.
// A operand (SRC0): x-chunk broadcast across all 16 M-rows, so every row of D
// holds the same 16 dot products; D VGPR0 lanes 0-15 = outputs for N=0..15.
template <bool RELU_X>
__device__ __forceinline__ float wave_dot16(const float* __restrict__ W,
                                            const float* __restrict__ x,
                                            int rowbase, int rows, int lane) {
  int row = rowbase + (lane & 15);
  if (row >= rows) row = rows - 1;               // clamp (duplicate read, write-masked later)
  const int khalf = (lane >> 4) << 1;            // 0 for lanes 0-15, 2 for lanes 16-31
  const float* wrow = W + (size_t)row * HIDDEN + khalf;
  const float* xk   = x + khalf;
  v8f c0 = {};
  v8f c1 = {};
#pragma unroll 4
  for (int k = 0; k < HIDDEN; k += 8) {
    v2f a0 = *(const v2f*)(xk + k);
    v2f a1 = *(const v2f*)(xk + k + 4);
    if (RELU_X) {
      a0[0] = fmaxf(a0[0], 0.f); a0[1] = fmaxf(a0[1], 0.f);
      a1[0] = fmaxf(a1[0], 0.f); a1[1] = fmaxf(a1[1], 0.f);
    }
    v2f w0 = *(const v2f*)(wrow + k);
    v2f w1 = *(const v2f*)(wrow + k + 4);
    // 8 args: (neg_a, A, neg_b, B, c_mod, C, reuse_a, reuse_b)
    c0 = __builtin_amdgcn_wmma_f32_16x16x4_f32(false, a0, false, w0, (short)0, c0, false, false);
    c1 = __builtin_amdgcn_wmma_f32_16x16x4_f32(false, a1, false, w1, (short)0, c1, false, false);
  }
  return c0[0] + c1[0];
}

// Fused GRU gate GEMVs: tiles [0,384) -> gi = w_ih @ relu(emb[token]) + b_ih
//                       tiles [384,768) -> gh = w_hh @ h + b_hh
__global__ __launch_bounds__(256) void gru_gemv_kernel(
    const int* __restrict__ token,
    const float* __restrict__ emb,
    const float* __restrict__ w_ih, const float* __restrict__ w_hh,
    const float* __restrict__ h,
    const float* __restrict__ b_ih, const float* __restrict__ b_hh,
    float* __restrict__ gi, float* __restrict__ gh) {
  const int wave = threadIdx.x >> 5;
  const int lane = threadIdx.x & 31;
  const int tile = blockIdx.x * WAVES_PER_BLOCK + wave;
  if (tile >= 2 * (3 * HIDDEN / 16)) return;

  const bool second = tile >= (3 * HIDDEN / 16);
  const int  rowbase = (second ? tile - (3 * HIDDEN / 16) : tile) << 4;

  float dot;
  if (second) {
    dot = wave_dot16<false>(w_hh, h, rowbase, 3 * HIDDEN, lane);
  } else {
    const float* x = emb + (size_t)token[0] * HIDDEN;   // relu applied on load
    dot = wave_dot16<true>(w_ih, x, rowbase, 3 * HIDDEN, lane);
  }

  const int orow = rowbase + lane;
  if (lane < 16) {
    if (second) gh[orow] = dot + b_hh[orow];
    else        gi[orow] = dot + b_ih[orow];
  }
}

// Elementwise GRU gates -> h_new (into ws and into d_out tail).
__global__ __launch_bounds__(256) void gru_gate_kernel(
    const float* __restrict__ gi, const float* __restrict__ gh,
    const float* __restrict__ h,
    float* __restrict__ h_new_ws, float* __restrict__ h_new_out) {
  const int i = blockIdx.x * blockDim.x + threadIdx.x;
  if (i >= HIDDEN) return;
  const float r = 1.f / (1.f + expf(-(gi[i] + gh[i])));
  const float z = 1.f / (1.f + expf(-(gi[HIDDEN + i] + gh[HIDDEN + i])));
  const float n = tanhf(gi[2 * HIDDEN + i] + r * gh[2 * HIDDEN + i]);
  const float hn = (1.f - z) * n + z * h[i];
  h_new_ws[i]  = hn;
  h_new_out[i] = hn;
}

// logits = w_out @ h_new + b_out  (50257 rows, last tile row-clamped)
__global__ __launch_bounds__(256) void logits_gemv_kernel(
    const float* __restrict__ w_out, const float* __restrict__ h_new,
    const float* __restrict__ b_out, float* __restrict__ logits, int rows) {
  const int wave = threadIdx.x >> 5;
  const int lane = threadIdx.x & 31;
  const int ntiles = (rows + 15) >> 4;
  const int tile = blockIdx.x * WAVES_PER_BLOCK + wave;
  if (tile >= ntiles) return;
  const int rowbase = tile << 4;
  const float dot = wave_dot16<false>(w_out, h_new, rowbase, rows, lane);
  const int orow = rowbase + lane;
  if (lane < 16 && orow < rows) logits[orow] = dot + b_out[orow];
}

// Single-block online log-sum-exp over the vocab: logZ = max + log(sum exp(l - max)).
__global__ __launch_bounds__(1024) void logsumexp_kernel(
    const float* __restrict__ logits, float* __restrict__ logZ) {
  __shared__ float sm[1024];
  __shared__ float ss[1024];
  const int tid = threadIdx.x;
  float m = -INFINITY, s = 0.f;
  for (int i = tid; i < VOCAB; i += 1024) {
    const float v = logits[i];
    if (v > m) { s = s * expf(m - v) + 1.f; m = v; }
    else       { s += expf(v - m); }
  }
  sm[tid] = m; ss[tid] = s;
  __syncthreads();
  for (int off = 512; off > 0; off >>= 1) {
    if (tid < off) {
      const float m1 = sm[tid], s1 = ss[tid];
      const float m2 = sm[tid + off], s2 = ss[tid + off];
      const float M = fmaxf(m1, m2);
      float S = 0.f;
      if (m1 != -INFINITY) S += s1 * expf(m1 - M);
      if (m2 != -INFINITY) S += s2 * expf(m2 - M);
      sm[tid] = M; ss[tid] = S;
    }
    __syncthreads();
  }
  if (tid == 0) logZ[0] = sm[0] + logf(ss[0]);
}

__global__ __launch_bounds__(256) void logsoftmax_write_kernel(
    const float* __restrict__ logits, const float* __restrict__ logZ,
    float* __restrict__ out) {
  const int i = blockIdx.x * blockDim.x + threadIdx.x;
  if (i < VOCAB) out[i] = logits[i] - logZ[0];
}

extern "C" void kernel_launch(void* const* d_in, const int* in_sizes, int n_in,
                              void* d_out, int out_size, void* d_ws, size_t ws_size,
                              hipStream_t stream) {
  const int*   token  = (const int*)  d_in[0];
  const float* hidden = (const float*)d_in[1];
  const float* emb    = (const float*)d_in[2];
  const float* w_ih   = (const float*)d_in[3];
  const float* w_hh   = (const float*)d_in[4];
  const float* b_ih   = (const float*)d_in[5];
  const float* b_hh   = (const float*)d_in[6];
  const float* w_out  = (const float*)d_in[7];
  const float* b_out  = (const float*)d_in[8];
  float* out = (float*)d_out;                 // [VOCAB] log-softmax, then [HIDDEN] h_new
  float* ws  = (float*)d_ws;

  float* GI = ws;                             // 3*HIDDEN
  float* GH = GI + 3 * HIDDEN;                // 3*HIDDEN
  float* HN = GH + 3 * HIDDEN;                // HIDDEN
  float* LG = HN + HIDDEN;                    // VOCAB
  float* LZ = LG + VOCAB;                     // 1

  // 1) gi and gh GEMVs fused into one launch: 768 tiles -> 96 blocks of 8 waves.
  gru_gemv_kernel<<<(2 * (3 * HIDDEN / 16) + WAVES_PER_BLOCK - 1) / WAVES_PER_BLOCK,
                    256, 0, stream>>>(token, emb, w_ih, w_hh, hidden, b_ih, b_hh, GI, GH);

  // 2) GRU gates -> h_new (ws + d_out tail).
  gru_gate_kernel<<<(HIDDEN + 255) / 256, 256, 0, stream>>>(GI, GH, hidden, HN, out + VOCAB);

  // 3) Output projection: 3142 tiles -> 393 blocks of 8 waves.
  {
    const int ntiles = (VOCAB + 15) / 16;
    logits_gemv_kernel<<<(ntiles + WAVES_PER_BLOCK - 1) / WAVES_PER_BLOCK,
                         256, 0, stream>>>(w_out, HN, b_out, LG, VOCAB);
  }

  // 4) log-sum-exp reduction, then 5) final subtraction.
  logsumexp_kernel<<<1, 1024, 0, stream>>>(LG, LZ);
  logsoftmax_write_kernel<<<(VOCAB + 255) / 256, 256, 0, stream>>>(LG, LZ, out);
}